// graphSAGE_56152402427956
// MI455X (gfx1250) — compile-verified
//
#include <hip/hip_runtime.h>

#define N_NODES 100000

typedef __attribute__((ext_vector_type(2))) float v2f;
typedef __attribute__((ext_vector_type(8))) float v8f;

// ---------------------------------------------------------------------------
// Degree count: cnt[dst[e]] += 1.0f
// ---------------------------------------------------------------------------
__global__ __launch_bounds__(256) void sage_count(const int* __restrict__ dst,
                                                  float* __restrict__ cnt,
                                                  int n_edges) {
    int e = blockIdx.x * 256 + threadIdx.x;
    if (e < n_edges) {
        atomicAdd(&cnt[dst[e]], 1.0f);
    }
}

// ---------------------------------------------------------------------------
// Edge scatter, vectorized: one thread per (edge, 4-channel quad).
//   agg[dst[e]*d + 4q .. +3] += feat[src[e]*d + 4q .. +3]
// d = 4 << qshift. One global_load_b128 gather + 4 consecutive-address
// f32 atomics (L2-resident: agg arrays << 192MB global L2); src/dst index
// loads amortized 4x vs scalar version.
// ---------------------------------------------------------------------------
__global__ __launch_bounds__(256) void sage_scatter4(const float4* __restrict__ feat4,
                                                     const int* __restrict__ src,
                                                     const int* __restrict__ dst,
                                                     float* __restrict__ agg,
                                                     int n_edges, int qshift) {
    long long t = (long long)blockIdx.x * 256 + threadIdx.x;
    int nq = 1 << qshift;                       // quads per node row
    long long total = (long long)n_edges << qshift;
    if (t >= total) return;
    int e = (int)(t >> qshift);
    int q = (int)(t & (nq - 1));

    float4 v = feat4[((long long)src[e] << qshift) + q];     // b128 gather
    float* p = agg + ((((long long)dst[e]) << qshift) + q) * 4;
    atomicAdd(p + 0, v.x);
    atomicAdd(p + 1, v.y);
    atomicAdd(p + 2, v.z);
    atomicAdd(p + 3, v.w);
}

// ---------------------------------------------------------------------------
// Fused SAGE linear layer via f32 WMMA:
//   out = act( (agg * 1/max(cnt,1)) @ Wl + bias + xin @ Wr )
// One wave32 computes one 16x16 tile of out using V_WMMA_F32_16X16X4_F32.
// A 16x4 tile: lane L holds row L%16, K pair {kh, kh+1} with kh = 2*(L/16).
// B 4x16 tile: lane L holds col L%16, same K pair.
// C/D: VGPR j = row j + 8*(L/16), col L%16.
// ---------------------------------------------------------------------------
__global__ __launch_bounds__(256) void sage_wmma_gemm(const float* __restrict__ agg,   // [N,K] sums
                                                      const float* __restrict__ cnt,   // [N]
                                                      const float* __restrict__ xin,   // [N,K]
                                                      const float* __restrict__ Wl,    // [K,C]
                                                      const float* __restrict__ bias,  // [C]
                                                      const float* __restrict__ Wr,    // [K,C]
                                                      float* __restrict__ out,         // [N,C]
                                                      int K, int C, int tn_shift,
                                                      int n_tiles, int do_relu) {
    int wave = threadIdx.x >> 5;
    int lane = threadIdx.x & 31;
    int tile = blockIdx.x * 8 + wave;
    if (tile >= n_tiles) return;

    int tm = tile >> tn_shift;              // node-tile index
    int tn = tile & ((1 << tn_shift) - 1);  // out-channel-tile index

    int l16   = lane & 15;
    int khalf = (lane >> 4) << 1;           // 0 or 2
    int row   = tm * 16 + l16;              // A row this lane holds
    int col   = tn * 16 + l16;              // B/C/D column this lane holds

    float inv = 1.0f / fmaxf(cnt[row], 1.0f);   // mean normalization for agg rows

    v8f c = {};

    const float* arow = xin + (long long)row * K + khalf;
    const float* grow = agg + (long long)row * K + khalf;

    // lin_r: xin @ Wr
    #pragma unroll 4
    for (int k = 0; k < K; k += 4) {
        v2f a = *(const v2f*)(arow + k);                 // 8B load: A[row][k+kh..kh+1]
        v2f b;
        b.x = Wr[(k + khalf + 0) * C + col];
        b.y = Wr[(k + khalf + 1) * C + col];
        c = __builtin_amdgcn_wmma_f32_16x16x4_f32(false, a, false, b,
                                                  (short)0, c, false, false);
    }
    // lin_l: (agg * inv) @ Wl
    #pragma unroll 4
    for (int k = 0; k < K; k += 4) {
        v2f a = *(const v2f*)(grow + k);
        a.x *= inv;
        a.y *= inv;
        v2f b;
        b.x = Wl[(k + khalf + 0) * C + col];
        b.y = Wl[(k + khalf + 1) * C + col];
        c = __builtin_amdgcn_wmma_f32_16x16x4_f32(false, a, false, b,
                                                  (short)0, c, false, false);
    }

    float bv = bias[col];
    int mbase = tm * 16 + ((lane >> 4) << 3);           // rows mbase..mbase+7
    #pragma unroll
    for (int j = 0; j < 8; ++j) {
        float v = c[j] + bv;
        if (do_relu) v = fmaxf(v, 0.0f);
        out[(long long)(mbase + j) * C + col] = v;
    }
}

// ---------------------------------------------------------------------------
// Launcher
// ---------------------------------------------------------------------------
extern "C" void kernel_launch(void* const* d_in, const int* in_sizes, int n_in,
                              void* d_out, int out_size, void* d_ws, size_t ws_size,
                              hipStream_t stream) {
    const float* x   = (const float*)d_in[0];   // [N,64]
    const int*   ei  = (const int*)  d_in[1];   // [2,E]
    const float* Wl1 = (const float*)d_in[2];   // [64,128]
    const float* bl1 = (const float*)d_in[3];   // [128]
    const float* Wr1 = (const float*)d_in[4];   // [64,128]
    const float* Wl2 = (const float*)d_in[5];   // [128,64]
    const float* bl2 = (const float*)d_in[6];   // [64]
    const float* Wr2 = (const float*)d_in[7];   // [128,64]
    float* out = (float*)d_out;

    const int E = in_sizes[1] / 2;
    const int* src = ei;
    const int* dst = ei + E;

    // Workspace layout (floats): cnt[N] | agg1[N*64] | h[N*128] | agg2[N*128]
    float* ws   = (float*)d_ws;
    float* cnt  = ws;
    float* agg1 = cnt + N_NODES;
    float* h    = agg1 + (size_t)N_NODES * 64;
    float* agg2 = h    + (size_t)N_NODES * 128;

    // Zero accumulators (cnt+agg1 are contiguous; h fully overwritten by GEMM1)
    hipMemsetAsync(cnt,  0, (size_t)N_NODES * 65  * sizeof(float), stream);
    hipMemsetAsync(agg2, 0, (size_t)N_NODES * 128 * sizeof(float), stream);

    // Degree counts (shared by both layers)
    sage_count<<<(E + 255) / 256, 256, 0, stream>>>(dst, cnt, E);

    // ---- Layer 1 ----
    {
        long long tot = (long long)E * 16;               // E * (64/4) quads
        sage_scatter4<<<(unsigned)((tot + 255) / 256), 256, 0, stream>>>(
            (const float4*)x, src, dst, agg1, E, /*qshift=*/4);

        int n_tiles = (N_NODES / 16) * (128 / 16);       // 6250 * 8 = 50000
        sage_wmma_gemm<<<(n_tiles + 7) / 8, 256, 0, stream>>>(
            agg1, cnt, x, Wl1, bl1, Wr1, h, /*K=*/64, /*C=*/128, /*tn_shift=*/3,
            n_tiles, /*relu=*/1);
    }

    // ---- Layer 2 ----
    {
        long long tot = (long long)E * 32;               // E * (128/4) quads
        sage_scatter4<<<(unsigned)((tot + 255) / 256), 256, 0, stream>>>(
            (const float4*)h, src, dst, agg2, E, /*qshift=*/5);

        int n_tiles = (N_NODES / 16) * (64 / 16);        // 6250 * 4 = 25000
        sage_wmma_gemm<<<(n_tiles + 7) / 8, 256, 0, stream>>>(
            agg2, cnt, h, Wl2, bl2, Wr2, out, /*K=*/128, /*C=*/64, /*tn_shift=*/2,
            n_tiles, /*relu=*/0);
    }
}